// BarLevelAttention_33200097198192
// MI455X (gfx1250) — compile-verified
//
#include <hip/hip_runtime.h>

typedef __attribute__((ext_vector_type(2))) float v2f;
typedef __attribute__((ext_vector_type(8))) float v8f;

#define S_LEN   2048
#define D_MODEL 512
#define N_HEADS 8
#define D_HEAD  64
#define SCALE   0.125f      // 1/sqrt(64)
#define NEGBIG  (-1e30f)

// ---------------------------------------------------------------------------
// 16-lane (half-wave) row reductions. wave32: rows 0-7 of a C-tile live in
// lanes 0-15, rows 8-15 in lanes 16-31, so width-16 shuffles reduce per row.
// ---------------------------------------------------------------------------
__device__ __forceinline__ float rowmax16(float v) {
  v = fmaxf(v, __shfl_xor(v, 8, 16));
  v = fmaxf(v, __shfl_xor(v, 4, 16));
  v = fmaxf(v, __shfl_xor(v, 2, 16));
  v = fmaxf(v, __shfl_xor(v, 1, 16));
  return v;
}
__device__ __forceinline__ float rowsum16(float v) {
  v += __shfl_xor(v, 8, 16);
  v += __shfl_xor(v, 4, 16);
  v += __shfl_xor(v, 2, 16);
  v += __shfl_xor(v, 1, 16);
  return v;
}

// CDNA5 async direct global->LDS copy (B128), tracked by ASYNCcnt.
// ldsDst = byte offset within the group segment (low 32 bits of generic ptr).
__device__ __forceinline__ void async_copy_b128(unsigned ldsDst,
                                                const void* gsrc) {
  asm volatile("global_load_async_to_lds_b128 %0, %1, off"
               :: "v"(ldsDst), "v"(gsrc)
               : "memory");
}
__device__ __forceinline__ void wait_async0() {
  asm volatile("s_wait_asynccnt 0x0" ::: "memory");
}

// ---------------------------------------------------------------------------
// bias4[r] = sum(bar_emb[r,:]^2), r = 0..3.  One block, 4 waves, 1 row each.
// ---------------------------------------------------------------------------
__global__ __launch_bounds__(128) void bar_bias_kernel(
    const float* __restrict__ emb, float* __restrict__ bias4) {
  const int lane = threadIdx.x & 31;
  const int wave = threadIdx.x >> 5;
  float s = 0.0f;
  for (int c = lane; c < D_MODEL; c += 32) {
    const float e = emb[wave * D_MODEL + c];
    s += e * e;
  }
  s += __shfl_xor(s, 16, 32);
  s += __shfl_xor(s, 8, 32);
  s += __shfl_xor(s, 4, 32);
  s += __shfl_xor(s, 2, 32);
  s += __shfl_xor(s, 1, 32);
  if (lane == 0) bias4[wave] = s;
}

// ---------------------------------------------------------------------------
// C[s,o] = sum_d A[s,d] * W[o,d] + b[o]   (A: 2048x512 rm, W: 512x512 rm)
// Block = 128 threads = 4 waves. Wave owns a 16(s) x 64(o) tile; k-step 4.
// headMajor: write C[o>>6][s][o&63] (for Q/K/V), else row-major [s][o].
// fp32 WMMA: full-precision match to the fp32 reference.
// ---------------------------------------------------------------------------
__global__ __launch_bounds__(128) void gemm_xWt(
    const float* __restrict__ A, const float* __restrict__ W,
    const float* __restrict__ bias, float* __restrict__ Out, int headMajor) {
  const int lane = threadIdx.x & 31;
  const int wave = threadIdx.x >> 5;
  const int lo = lane & 15;
  const int hi = lane >> 4;
  const int K0 = hi * 2;                        // A/B K offset within k-step
  const int sBase = blockIdx.x * 16;
  const int oBase = blockIdx.y * 256 + wave * 64;

  v8f acc[4] = {};
  const float* arow  = A + (size_t)(sBase + lo) * D_MODEL;   // A: M = lo
  const float* wrow0 = W + (size_t)(oBase + lo) * D_MODEL;   // B: N = lo

  for (int d = 0; d < D_MODEL; d += 4) {
    const v2f a = *(const v2f*)(arow + d + K0);              // A[M][d+K0..+1]
#pragma unroll
    for (int nt = 0; nt < 4; ++nt) {
      const v2f b = *(const v2f*)(wrow0 + (size_t)nt * 16 * D_MODEL + d + K0);
      acc[nt] = __builtin_amdgcn_wmma_f32_16x16x4_f32(
          false, a, false, b, (short)0, acc[nt], false, false);
    }
  }

#pragma unroll
  for (int nt = 0; nt < 4; ++nt) {
    const int o = oBase + nt * 16 + lo;
    const float bv = bias[o];
#pragma unroll
    for (int i = 0; i < 8; ++i) {
      const int row = sBase + i + 8 * hi;                    // C: M = i + 8*hi
      const float val = acc[nt][i] + bv;
      if (headMajor) {
        const int hh = o >> 6, dh = o & 63;
        Out[((size_t)hh * S_LEN + row) * D_HEAD + dh] = val;
      } else {
        Out[(size_t)row * D_MODEL + o] = val;
      }
    }
  }
}

// ---------------------------------------------------------------------------
// Fused dual-stream flash attention.
// Grid (S/128, H); block = 256 threads = 8 waves; wave owns 16 queries.
// K/V tiles are double-buffered in LDS and filled with
// global_load_async_to_lds_b128 (ASYNCcnt), overlapping the next tile's
// fill with the current tile's 48 WMMAs + softmax.
// Per 16-key tile: one QK^T (16 WMMAs), then local (bar-masked) and global
// (bias-shifted) online softmaxes each followed by a P*V (16 WMMAs each).
// P goes C-layout -> LDS -> A-layout (per-wave buffer; LDS is in-order).
// ---------------------------------------------------------------------------
__global__ __launch_bounds__(256) void bar_attn(
    const float* __restrict__ Q, const float* __restrict__ K,
    const float* __restrict__ V, const int* __restrict__ barpos,
    const unsigned char* __restrict__ amask, const float* __restrict__ bias4,
    const float* __restrict__ gate, float* __restrict__ Comb) {
  __shared__ __align__(16) float Kt[2][16 * D_HEAD];
  __shared__ __align__(16) float Vt[2][16 * D_HEAD];
  __shared__ __align__(16) float Pt[8 * 256];

  const int t = threadIdx.x;
  const int lane = t & 31;
  const int wave = t >> 5;
  const int lo = lane & 15;
  const int hi = lane >> 4;
  const int K0 = hi * 2;
  const int h = blockIdx.y;
  const int qBase = blockIdx.x * 128 + wave * 16;

  const float* Qh = Q + (size_t)h * S_LEN * D_HEAD;
  const float* Kh = K + (size_t)h * S_LEN * D_HEAD;
  const float* Vh = V + (size_t)h * S_LEN * D_HEAD;

  // Q tile as A-fragments: qa[j] = Q[qBase+lo][4j+K0 .. 4j+K0+1]
  v2f qa[16];
#pragma unroll
  for (int j = 0; j < 16; ++j)
    qa[j] = *(const v2f*)(Qh + (size_t)(qBase + lo) * D_HEAD + 4 * j + K0);

  int barQ[8];
#pragma unroll
  for (int i = 0; i < 8; ++i) barQ[i] = barpos[qBase + i + 8 * hi];

  const float g = 1.0f / (1.0f + __expf(-gate[h]));

  float mL[8], lL[8], mG[8], lG[8];
  v8f accL[4] = {};
  v8f accG[4] = {};
#pragma unroll
  for (int i = 0; i < 8; ++i) {
    mL[i] = NEGBIG; lL[i] = 0.0f;
    mG[i] = NEGBIG; lG[i] = 0.0f;
  }

  float* myP = Pt + wave * 256;

  // LDS byte offsets for this thread's 16B staging slot in each buffer.
  const unsigned ldsK0 = (unsigned)(uintptr_t)&Kt[0][0] + (unsigned)t * 16u;
  const unsigned ldsK1 = (unsigned)(uintptr_t)&Kt[1][0] + (unsigned)t * 16u;
  const unsigned ldsV0 = (unsigned)(uintptr_t)&Vt[0][0] + (unsigned)t * 16u;
  const unsigned ldsV1 = (unsigned)(uintptr_t)&Vt[1][0] + (unsigned)t * 16u;

  // issue async fill of one 16x64 K tile + V tile (4 KB each, 16 B/thread)
  auto issue_tile = [&](int buf, int kt2) {
    const float4* kp = (const float4*)(Kh + (size_t)kt2 * 16 * D_HEAD) + t;
    const float4* vp = (const float4*)(Vh + (size_t)kt2 * 16 * D_HEAD) + t;
    async_copy_b128(buf ? ldsK1 : ldsK0, kp);
    async_copy_b128(buf ? ldsV1 : ldsV0, vp);
  };

  issue_tile(0, 0);

  for (int kt = 0; kt < S_LEN / 16; ++kt) {
    const int kBase = kt * 16;
    const int cur = kt & 1;

    wait_async0();        // our async fills of buffer `cur` hit LDS
    __syncthreads();      // everyone's fills done; everyone off buffer cur^1
    if (kt + 1 < S_LEN / 16) issue_tile(cur ^ 1, kt + 1);
    if (kt + 2 < S_LEN / 16) {                  // deepen the prefetch pipe
      __builtin_prefetch(Kh + (size_t)(kBase + 32) * D_HEAD, 0, 0);
      __builtin_prefetch(Vh + (size_t)(kBase + 32) * D_HEAD, 0, 0);
    }

    const float* Ktc = &Kt[cur][0];
    const float* Vtc = &Vt[cur][0];

    const int kcol = kBase + lo;                // this lane's key column
    const int barK = barpos[kcol];
    const float biasK = bias4[barK & 3];
    const bool padK = (amask[kcol] != 0);

    // scores S = Q * K^T : B-frag b.x = K[kBase+lo][4j+K0]
    v8f sc = {};
#pragma unroll
    for (int j = 0; j < 16; ++j) {
      const v2f b = *(const v2f*)(Ktc + lo * D_HEAD + 4 * j + K0);
      sc = __builtin_amdgcn_wmma_f32_16x16x4_f32(
          false, qa[j], false, b, (short)0, sc, false, false);
    }

    // ---- local stream: bar-masked softmax ----
#pragma unroll
    for (int i = 0; i < 8; ++i) {
      const float s = sc[i] * SCALE;
      const float sl = (padK && (barK == barQ[i])) ? s : NEGBIG;
      const float mNew = fmaxf(mL[i], rowmax16(sl));
      const float corr = __expf(mL[i] - mNew);
      const float p = __expf(sl - mNew);
      lL[i] = lL[i] * corr + rowsum16(p);
      mL[i] = mNew;
#pragma unroll
      for (int nt = 0; nt < 4; ++nt) accL[nt][i] *= corr;
      myP[(i + 8 * hi) * 16 + lo] = p;          // C-layout -> LDS
    }
#pragma unroll
    for (int nt = 0; nt < 4; ++nt) {
#pragma unroll
      for (int j = 0; j < 4; ++j) {
        const v2f a = *(const v2f*)(myP + lo * 16 + 4 * j + K0);  // A-layout
        v2f b;
        b.x = Vtc[(4 * j + K0) * D_HEAD + nt * 16 + lo];
        b.y = Vtc[(4 * j + K0 + 1) * D_HEAD + nt * 16 + lo];
        accL[nt] = __builtin_amdgcn_wmma_f32_16x16x4_f32(
            false, a, false, b, (short)0, accL[nt], false, false);
      }
    }

    // ---- global stream: bias-shifted softmax ----
#pragma unroll
    for (int i = 0; i < 8; ++i) {
      const float s = sc[i] * SCALE;
      const float sg = padK ? (s + biasK) : NEGBIG;
      const float mNew = fmaxf(mG[i], rowmax16(sg));
      const float corr = __expf(mG[i] - mNew);
      const float p = __expf(sg - mNew);
      lG[i] = lG[i] * corr + rowsum16(p);
      mG[i] = mNew;
#pragma unroll
      for (int nt = 0; nt < 4; ++nt) accG[nt][i] *= corr;
      myP[(i + 8 * hi) * 16 + lo] = p;
    }
#pragma unroll
    for (int nt = 0; nt < 4; ++nt) {
#pragma unroll
      for (int j = 0; j < 4; ++j) {
        const v2f a = *(const v2f*)(myP + lo * 16 + 4 * j + K0);
        v2f b;
        b.x = Vtc[(4 * j + K0) * D_HEAD + nt * 16 + lo];
        b.y = Vtc[(4 * j + K0 + 1) * D_HEAD + nt * 16 + lo];
        accG[nt] = __builtin_amdgcn_wmma_f32_16x16x4_f32(
            false, a, false, b, (short)0, accG[nt], false, false);
      }
    }
  }

  // finalize: gated combine, write [s][h*64+dh]
#pragma unroll
  for (int i = 0; i < 8; ++i) {
    const float il = 1.0f / lL[i];
    const float ig = 1.0f / lG[i];
    const int row = qBase + i + 8 * hi;
#pragma unroll
    for (int nt = 0; nt < 4; ++nt) {
      const float val = g * accL[nt][i] * il + (1.0f - g) * accG[nt][i] * ig;
      Comb[(size_t)row * D_MODEL + h * D_HEAD + nt * 16 + lo] = val;
    }
  }
}

// ---------------------------------------------------------------------------
extern "C" void kernel_launch(void* const* d_in, const int* in_sizes, int n_in,
                              void* d_out, int out_size, void* d_ws,
                              size_t ws_size, hipStream_t stream) {
  const float* X      = (const float*)d_in[0];
  const int*   barpos = (const int*)d_in[1];
  const unsigned char* amask = (const unsigned char*)d_in[2];
  const float* Wq = (const float*)d_in[3];
  const float* bq = (const float*)d_in[4];
  const float* Wk = (const float*)d_in[5];
  const float* bk = (const float*)d_in[6];
  const float* Wv = (const float*)d_in[7];
  const float* bv = (const float*)d_in[8];
  const float* Wo = (const float*)d_in[9];
  const float* bo = (const float*)d_in[10];
  const float* bar_emb = (const float*)d_in[11];
  const float* gate = (const float*)d_in[12];
  float* out = (float*)d_out;

  // workspace: q, k, v (head-major), combined, bias4  (~16 MB + 16 B)
  float* ws = (float*)d_ws;
  const size_t N = (size_t)S_LEN * D_MODEL;
  float* q     = ws;
  float* k     = ws + N;
  float* v     = ws + 2 * N;
  float* comb  = ws + 3 * N;
  float* bias4 = ws + 4 * N;

  bar_bias_kernel<<<1, 128, 0, stream>>>(bar_emb, bias4);

  const dim3 gg(S_LEN / 16, D_MODEL / 256);
  gemm_xWt<<<gg, 128, 0, stream>>>(X, Wq, bq, q, 1);
  gemm_xWt<<<gg, 128, 0, stream>>>(X, Wk, bk, k, 1);
  gemm_xWt<<<gg, 128, 0, stream>>>(X, Wv, bv, v, 1);

  bar_attn<<<dim3(S_LEN / 128, N_HEADS), 256, 0, stream>>>(
      q, k, v, barpos, amask, bias4, gate, comb);

  gemm_xWt<<<gg, 128, 0, stream>>>(comb, Wo, bo, out, 0);

  (void)in_sizes; (void)n_in; (void)out_size; (void)ws_size;
}